// TorsionLayer_82557861363801
// MI455X (gfx1250) — compile-verified
//
#include <hip/hip_runtime.h>

typedef float v4f __attribute__((ext_vector_type(4)));

#define TORSION_PI     3.14159265358979323846f
#define TORSION_FACTOR 57.295779513082320877f /* 180/pi */

// Per-point math:
//   phi = acos(z)
//   sin(phi) = sqrt(1 - z^2)  (exact identity, acos range [0,pi])
//   rat = clamp(y * rsqrt(1 - z^2), -1, 1)
//   psi = asin(rat); if (x<0) psi = copysign(pi, y) - psi
//   out = (-phi, psi) * 180/pi
__device__ __forceinline__ void torsion_point(float x, float y, float z,
                                              float& o_phi, float& o_psi) {
  float t   = fmaxf(__builtin_fmaf(-z, z, 1.0f), 0.0f); // sin^2(phi), clamped vs rounding
  float rat = y * __builtin_amdgcn_rsqf(t);             // y / sin(phi) via v_rsq_f32
  rat       = fminf(1.0f, fmaxf(-1.0f, rat));
  float phi = acosf(z);
  float psi = asinf(rat);
  float alt = __builtin_copysignf(TORSION_PI, y) - psi; // (-pi-psi) if y<0 else (pi-psi)
  psi       = (x < 0.0f) ? alt : psi;
  o_phi = -phi * TORSION_FACTOR;
  o_psi =  psi * TORSION_FACTOR;
}

__global__ __launch_bounds__(256)
void torsion_kernel(const float* __restrict__ in, float* __restrict__ out,
                    long long n_points) {
  long long tid = (long long)blockIdx.x * blockDim.x + threadIdx.x;
  long long p   = tid * 4;              // 4 points per lane

  if (p + 4 <= n_points) {
    const v4f* in4  = (const v4f*)in;   // 4 pts = 12 floats = 3 x b128
    v4f*       out4 = (v4f*)out;        // 4 pts =  8 floats = 2 x b128

    // Streaming input: prefetch ~24 KiB ahead (lowers to global_prefetch_b8).
    __builtin_prefetch(in + p * 3 + 6144, 0, 1);

    // Read-once data: non-temporal (TH=NT) b128 loads, coalesced across wave32.
    v4f a = __builtin_nontemporal_load(&in4[tid * 3 + 0]); // x0 y0 z0 x1
    v4f b = __builtin_nontemporal_load(&in4[tid * 3 + 1]); // y1 z1 x2 y2
    v4f c = __builtin_nontemporal_load(&in4[tid * 3 + 2]); // z2 x3 y3 z3

    float o0, o1, o2, o3, q0, q1, q2, q3;
    torsion_point(a[0], a[1], a[2], o0, q0);
    torsion_point(a[3], b[0], b[1], o1, q1);
    torsion_point(b[2], b[3], c[0], o2, q2);
    torsion_point(c[1], c[2], c[3], o3, q3);

    v4f r0 = {o0, q0, o1, q1};
    v4f r1 = {o2, q2, o3, q3};
    __builtin_nontemporal_store(r0, &out4[tid * 2 + 0]);
    __builtin_nontemporal_store(r1, &out4[tid * 2 + 1]);
  } else {
    // Scalar tail (not hit for 4096*2048, kept for generality).
    for (long long i = p; i < n_points; ++i) {
      float o, q;
      torsion_point(in[i * 3 + 0], in[i * 3 + 1], in[i * 3 + 2], o, q);
      out[i * 2 + 0] = o;
      out[i * 2 + 1] = q;
    }
  }
}

extern "C" void kernel_launch(void* const* d_in, const int* in_sizes, int n_in,
                              void* d_out, int out_size, void* d_ws, size_t ws_size,
                              hipStream_t stream) {
  (void)n_in; (void)out_size; (void)d_ws; (void)ws_size;
  const float* in  = (const float*)d_in[0];
  float*       out = (float*)d_out;
  long long n_points = (long long)in_sizes[0] / 3;  // [B,N,3] flattened
  long long nvec     = (n_points + 3) / 4;          // 4 points per thread
  const int block    = 256;                         // 8 wave32s per block
  unsigned  grid     = (unsigned)((nvec + block - 1) / block);
  torsion_kernel<<<dim3(grid), dim3(block), 0, stream>>>(in, out, n_points);
}